// OSTFRecurrentFieldWorldModelV32_4312147165459
// MI455X (gfx1250) — compile-verified
//
#include <hip/hip_runtime.h>

// CDNA5 / gfx1250, wave32.
// Output (6, B, M, 32, 2) f32 = 201 MB  ->  store-bandwidth bound (~8.6 us @ 23.3 TB/s).
// priors 0,3,4 == broadcast(last_visible)  (median+layout == last_visible; gamma == 0)
// prior  1     == broadcast(last_observed)
// prior  2     == last_visible + vel_visible * t
// prior  5     == last_observed + 0.25*vel * t
//
// Each wave handles 16 agents via V_WMMA_F32_16X16X4_F32:
//   D(16x16) = A(16x4: per-agent [base, vel] per coord) x B(4x16: time basis [1, t])
// D columns n = h*2+c  ->  storing accumulator VGPR v writes agent (tile+v)'s 16
// contiguous floats in lanes 0-15 and agent (tile+v+8)'s in lanes 16-31: every
// b32 store covers two full 64B cachelines. Non-temporal stores (output > L2).

typedef __attribute__((ext_vector_type(2))) float v2f;
typedef __attribute__((ext_vector_type(8))) float v8f;

#define OBS_LEN 8

__device__ __forceinline__ float fixv(float x) {
    if (__builtin_isnan(x)) return 0.0f;
    if (__builtin_isinf(x)) return x > 0.0f ? 1.0e6f : -1.0e6f;
    return x;
}

__device__ __forceinline__ v8f wmma4(v2f a, v2f b) {
    v8f c = {};
    // (neg_a, A, neg_b, B, c_mod, C, reuse_a, reuse_b)
    return __builtin_amdgcn_wmma_f32_16x16x4_f32(false, a, false, b, (short)0, c,
                                                 false, false);
}

__global__ __launch_bounds__(256) void
ostf_priors_wmma_kernel(const float* __restrict__ obs,
                        const unsigned char* __restrict__ vis,
                        float* __restrict__ out, int nAgents) {
    const int lane = threadIdx.x & 31;
    const int wave = threadIdx.x >> 5;
    const int tile = blockIdx.x * 8 + wave;       // 16 agents per wave-tile
    if (tile * 16 >= nAgents) return;             // wave-uniform: EXEC stays full

    const int aLocal = lane & 15;                 // agent within tile (A row / B col)
    const int half   = lane >> 4;                 // 0: x / K0-K1   1: y / K2-K3
    const long agent = (long)tile * 16 + aLocal;

    // ---- per-lane stats for (agent, coordinate=half) ----------------------
    const float* pp = obs + agent * (OBS_LEN * 2) + half;
    float q[OBS_LEN];
#pragma unroll
    for (int t = 0; t < OBS_LEN; ++t) q[t] = fixv(pp[2 * t]);

    // 8 visibility bytes as one b64 load
    const unsigned long long vb =
        *(const unsigned long long*)(vis + agent * OBS_LEN);

    int   last_i = -1, prev_i = -1;
    float last_v = 0.0f, prev_v = 0.0f;
#pragma unroll
    for (int t = 0; t < OBS_LEN; ++t) {
        if ((vb >> (8 * t)) & 0xffull) {
            prev_i = last_i; prev_v = last_v;
            last_i = t;      last_v = q[t];
        }
    }
    const bool  hasv  = last_i >= 0;
    const float lv    = hasv ? last_v : q[OBS_LEN - 1];       // last_visible
    float vvel = 0.0f;                                        // visible velocity
    if (prev_i >= 0) vvel = (last_v - prev_v) / (float)(last_i - prev_i);
    const float lo = q[OBS_LEN - 1];                          // last_observed
    const float vq = 0.25f * (q[OBS_LEN - 1] - q[OBS_LEN - 2]);

    // A fragments: lanes 0-15 hold K0,K1 for M=lane; lanes 16-31 hold K2,K3
    const v2f a_lv = {lv, vvel};
    const v2f a_lo = {lo, vq};

    // B fragments (time basis 4x16): col n = h*2 + c; K0/K1 gate c==0, K2/K3 gate c==1.
    // VGPR0: lanes0-15 K=0(n=lane), lanes16-31 K=2(n=lane-16); VGPR1: K=1 / K=3.
    const int   n   = lane & 15;
    const int   cn  = n & 1;
    const int   hl  = n >> 1;                     // horizon within block (0..7)
    const float sel = (cn == half) ? 1.0f : 0.0f;
    const v2f   b0  = {sel, 0.0f};                // t == 0  ->  pure broadcast

    const v8f bc_lv = wmma4(a_lv, b0);            // priors 0,3,4
    const v8f bc_lo = wmma4(a_lo, b0);            // prior 1

    const long PR     = (long)nAgents * 64;       // floats per prior
    const int  rowAdd = half * 8;                 // lanes>=16 hold rows M=v+8
    float* baseT = out + ((long)tile * 16 + rowAdd) * 64 + n;

    // ---- broadcast priors: same data at all 4 horizon blocks --------------
#pragma unroll
    for (int j = 0; j < 4; ++j) {
#pragma unroll
        for (int v = 0; v < 8; ++v) {
            const long off = (long)j * 16 + (long)v * 64;
            const float x  = bc_lv[v];
            __builtin_nontemporal_store(x,        baseT + 0 * PR + off);
            __builtin_nontemporal_store(x,        baseT + 3 * PR + off);
            __builtin_nontemporal_store(x,        baseT + 4 * PR + off);
            __builtin_nontemporal_store(bc_lo[v], baseT + 1 * PR + off);
        }
    }

    // ---- extrapolation priors: t = j*8 + hl + 1 ---------------------------
#pragma unroll
    for (int j = 0; j < 4; ++j) {
        const v2f bt = {sel, sel * (float)(j * 8 + hl + 1)};
        const v8f d2 = wmma4(a_lv, bt);           // prior 2
        const v8f d5 = wmma4(a_lo, bt);           // prior 5
#pragma unroll
        for (int v = 0; v < 8; ++v) {
            const long off = (long)j * 16 + (long)v * 64;
            __builtin_nontemporal_store(d2[v], baseT + 2 * PR + off);
            __builtin_nontemporal_store(d5[v], baseT + 5 * PR + off);
        }
    }
}

extern "C" void kernel_launch(void* const* d_in, const int* in_sizes, int n_in,
                              void* d_out, int out_size, void* d_ws, size_t ws_size,
                              hipStream_t stream) {
    const float*         obs = (const float*)d_in[0];
    const unsigned char* vis = (const unsigned char*)d_in[1];  // jax bool: 1 byte
    float* out = (float*)d_out;

    const int nAgents = in_sizes[1] / OBS_LEN;    // B*M = 131072
    const int tiles   = nAgents / 16;             // 16 agents per wave
    const int blocks  = (tiles + 7) / 8;          // 8 waves (256 threads) per block

    hipLaunchKernelGGL(ostf_priors_wmma_kernel, dim3(blocks), dim3(256), 0, stream,
                       obs, vis, out, nAgents);
}